// DraftSampler_56229711839575
// MI455X (gfx1250) — compile-verified
//
#include <hip/hip_runtime.h>
#include <hip/hip_bf16.h>
#include <stdint.h>

typedef float v4f __attribute__((ext_vector_type(4)));
typedef int v4i __attribute__((ext_vector_type(4)));

#define SAMPLER_EPS 1e-10f
#define BLOCK 512
#define TILE (BLOCK * 4) /* 2048 floats per stage per tensor */
#define NBUF 4           /* 4-deep async pipeline */

#define AS1 __attribute__((address_space(1)))
#define AS3 __attribute__((address_space(3)))

// CPol: TH[2:0]; TH_NT = 1 (stream >> L2, zero reuse -> non-temporal)
#define CPOL_NT 1

// ---- CDNA5 async global->LDS copy (B128) -----------------------------------
#if __has_builtin(__builtin_amdgcn_global_load_async_to_lds_b128)
#define ASYNC_B128(gp, lp)                                                     \
  __builtin_amdgcn_global_load_async_to_lds_b128(                              \
      (AS1 v4i*)(uintptr_t)(gp), (AS3 v4i*)(uint32_t)(uintptr_t)(lp), 0,       \
      CPOL_NT)
#else
#define ASYNC_B128(gp, lp)                                                     \
  do {                                                                         \
    unsigned long long _ga = (unsigned long long)(uintptr_t)(gp);              \
    uint32_t _la = (uint32_t)(uintptr_t)(lp);                                  \
    asm volatile("global_load_async_to_lds_b128 %0, %1, off th:TH_LOAD_NT" ::  \
                     "v"(_la),                                                 \
                 "v"(_ga)                                                      \
                 : "memory");                                                  \
  } while (0)
#endif

#if __has_builtin(__builtin_amdgcn_s_wait_asynccnt)
#define WAIT_ASYNC(n) __builtin_amdgcn_s_wait_asynccnt(n)
#else
#define WAIT_ASYNC(n) asm volatile("s_wait_asynccnt %0" ::"i"(n) : "memory")
#endif
// ----------------------------------------------------------------------------

__global__ __launch_bounds__(BLOCK, 1) void DraftSampler_56229711839575_kernel(
    const float* __restrict__ logits, const float* __restrict__ temps,
    const float* __restrict__ noise, float* __restrict__ out, int V) {
  __shared__ v4f bufL[NBUF][BLOCK];
  __shared__ v4f bufN[NBUF][BLOCK];
  __shared__ float sVal[BLOCK / 32];
  __shared__ int sIdx[BLOCK / 32];

  const int row = blockIdx.x;
  const int tid = threadIdx.x;

  const float temp = temps[row];       // uniform per block -> scalar load
  const bool greedy = (temp == 0.0f);  // uniform branch, no divergence
  const float invT = greedy ? 1.0f : (1.0f / temp);

  const float* __restrict__ lrow = logits + (size_t)row * (size_t)V;
  const float* __restrict__ nrow = noise + (size_t)row * (size_t)V;

  float bestV = -__builtin_inff();
  int bestI = 0x7fffffff;

  const int V4 = V & ~3;
  const int nStages = (V4 + TILE - 1) / TILE;
  const int lane4 = tid * 4;

  // Issue one stage's async copies. Only the final stage can be partial
  // (uniform condition), so steady state has no per-lane clamp VALU.
  auto issueStage = [&](int s) {
    const int b = s & (NBUF - 1);
    const int base = s * TILE;
    if (base + TILE <= V4) {
      ASYNC_B128(lrow + base + lane4, &bufL[b][tid]);
      if (!greedy) ASYNC_B128(nrow + base + lane4, &bufN[b][tid]);
    } else {
      const int g = base + lane4;
      const int cg = (g < V4) ? g : 0;  // clamp OOB lanes (masked later)
      ASYNC_B128(lrow + cg, &bufL[b][tid]);
      if (!greedy) ASYNC_B128(nrow + cg, &bufN[b][tid]);
    }
  };

  // Prologue: up to 3 stages in flight.
  if (nStages > 0) issueStage(0);
  if (nStages > 1) issueStage(1);
  if (nStages > 2) issueStage(2);

  for (int s = 0; s < nStages; ++s) {
    if (s + 3 < nStages) issueStage(s + 3);

    // Stages pending after stage s completes (async ops retire in order).
    const int rem0 = nStages - s - 1;
    const int rem = rem0 > 3 ? 3 : rem0;
    if (greedy) {  // 1 async op per stage
      switch (rem) {
        case 3: WAIT_ASYNC(3); break;
        case 2: WAIT_ASYNC(2); break;
        case 1: WAIT_ASYNC(1); break;
        default: WAIT_ASYNC(0); break;
      }
    } else {  // 2 async ops per stage
      switch (rem) {
        case 3: WAIT_ASYNC(6); break;
        case 2: WAIT_ASYNC(4); break;
        case 1: WAIT_ASYNC(2); break;
        default: WAIT_ASYNC(0); break;
      }
    }

    const int b = s & (NBUF - 1);
    const int gidx = s * TILE + lane4;
    if (gidx < V4) {  // each lane reads only its own LDS slot: no barrier
      const v4f lv = bufL[b][tid];  // ds_load_b128
      if (greedy) {
#pragma unroll
        for (int j = 0; j < 4; ++j) {
          const float v = lv[j];
          const int i = gidx + j;
          if (v > bestV || (v == bestV && i < bestI)) { bestV = v; bestI = i; }
        }
      } else {
        const v4f nv = bufN[b][tid];
#pragma unroll
        for (int j = 0; j < 4; ++j) {
          // argmax( softmax(x/t)/(n+eps) ) == argmax( x/t - log(n+eps) )
          const float v =
              __builtin_fmaf(lv[j], invT, -__logf(nv[j] + SAMPLER_EPS));
          const int i = gidx + j;
          if (v > bestV || (v == bestV && i < bestI)) { bestV = v; bestI = i; }
        }
      }
    }
  }

  // Scalar tail if V % 4 != 0 (not hit for V=128000, kept for generality).
  for (int i = V4 + tid; i < V; i += BLOCK) {
    const float l = lrow[i];
    const float v =
        greedy ? l : __builtin_fmaf(l, invT, -__logf(nrow[i] + SAMPLER_EPS));
    if (v > bestV || (v == bestV && i < bestI)) { bestV = v; bestI = i; }
  }

  // wave32 argmax reduction (first-index tie-break)
#pragma unroll
  for (int off = 16; off >= 1; off >>= 1) {
    const float ov = __shfl_xor(bestV, off, 32);
    const int oi = __shfl_xor(bestI, off, 32);
    if (ov > bestV || (ov == bestV && oi < bestI)) { bestV = ov; bestI = oi; }
  }
  const int wave = tid >> 5, lane = tid & 31;
  if (lane == 0) { sVal[wave] = bestV; sIdx[wave] = bestI; }
  __syncthreads();
  if (wave == 0) {
    const int nw = BLOCK / 32;
    bestV = (lane < nw) ? sVal[lane] : -__builtin_inff();
    bestI = (lane < nw) ? sIdx[lane] : 0x7fffffff;
#pragma unroll
    for (int off = 16; off >= 1; off >>= 1) {
      const float ov = __shfl_xor(bestV, off, 32);
      const int oi = __shfl_xor(bestI, off, 32);
      if (ov > bestV || (ov == bestV && oi < bestI)) { bestV = ov; bestI = oi; }
    }
    if (lane == 0) out[row] = (float)bestI;
  }
}

extern "C" void kernel_launch(void* const* d_in, const int* in_sizes, int n_in,
                              void* d_out, int out_size, void* d_ws,
                              size_t ws_size, hipStream_t stream) {
  (void)n_in;
  (void)d_ws;
  (void)ws_size;
  (void)out_size;
  const float* logits = (const float*)d_in[0];
  const float* temps = (const float*)d_in[1];
  const float* noise = (const float*)d_in[2];
  float* out = (float*)d_out;

  const int B = in_sizes[1];
  if (B <= 0) return;
  const int V = in_sizes[0] / B;

  DraftSampler_56229711839575_kernel<<<B, BLOCK, 0, stream>>>(logits, temps,
                                                              noise, out, V);
}